// Sylo_44787918962918
// MI455X (gfx1250) — compile-verified
//
#include <hip/hip_runtime.h>
#include <hip/hip_bf16.h>

// dims: N batch, O out_channels, CI in_channels, H=W vertices, R rank
#define N_ 8
#define O_ 32
#define CI_ 16
#define H_ 512
#define R_ 8

typedef float v2f __attribute__((ext_vector_type(2)));
typedef float v8f __attribute__((ext_vector_type(8)));

// ---------------------------------------------------------------------------
// Kernel 1: P[n,o,h,w] = sum_c X[n,c,h,w] * (Lc[o,c] @ L[o,c]^T)[h,w]
// One wave handles one 16x16 (h,w) tile for one output channel o.
// The 8 waves of a block share the same (h,w) tile (different o) so the X
// tile loads hit the WGP cache 8x.  T tile built with V_WMMA_F32_16X16X4_F32
// (K=8 -> two chained WMMAs), X contraction done with VALU FMAs against the
// WMMA accumulator layout (M = reg + 8*half, N = lane&15).
// ---------------------------------------------------------------------------
__global__ __launch_bounds__(256) void sylo_main(const float* __restrict__ X,
                                                 const float* __restrict__ L,
                                                 const float* __restrict__ S,
                                                 float* __restrict__ P) {
  const int lane = threadIdx.x & 31;
  const int wave = threadIdx.x >> 5;   // 0..7
  const int hh   = lane >> 4;          // half-wave: 0/1
  const int lcol = lane & 15;          // A: M row / B: N col / C,D: N col

  const int bid = blockIdx.x;          // 4096 blocks
  const int wt  = bid & 31;
  const int ht  = (bid >> 5) & 31;
  const int og  = bid >> 10;           // 0..3
  const int o   = og * 8 + wave;       // 0..31

  const int h0 = ht * 16;
  const int w0 = wt * 16;
  const int k0 = 2 * hh;               // A/B fragment: K = reg + 2*half

  float p[N_][8];
#pragma unroll
  for (int n = 0; n < N_; ++n)
#pragma unroll
    for (int k = 0; k < 8; ++k) p[n][k] = 0.0f;

  const float* Lo = L + (size_t)o * CI_ * H_ * R_;
  const float* So = S + (size_t)o * CI_ * R_;

  for (int c = 0; c < CI_; ++c) {
    const float* Lc = Lo + (size_t)c * H_ * R_;
    const float* Sc = So + (size_t)c * R_;

    // A fragments: rows h0..h0+15 of Lc (sign-coupled), K = {k0,k0+1} and +4
    v2f a0 = *(const v2f*)(Lc + (size_t)(h0 + lcol) * R_ + k0);
    v2f a1 = *(const v2f*)(Lc + (size_t)(h0 + lcol) * R_ + 4 + k0);
    v2f s0 = *(const v2f*)(Sc + k0);
    v2f s1 = *(const v2f*)(Sc + 4 + k0);
    a0 *= s0;
    a1 *= s1;
    // B fragments: B[k][n] = L[w0+n, k] (T = Lc * L^T)
    v2f b0 = *(const v2f*)(Lc + (size_t)(w0 + lcol) * R_ + k0);
    v2f b1 = *(const v2f*)(Lc + (size_t)(w0 + lcol) * R_ + 4 + k0);

    v8f t = {};
    t = __builtin_amdgcn_wmma_f32_16x16x4_f32(false, a0, false, b0,
                                              (short)0, t, false, false);
    t = __builtin_amdgcn_wmma_f32_16x16x4_f32(false, a1, false, b1,
                                              (short)0, t, false, false);

    // p[n] += X[n,c,tile] .* t   (C/D layout: row = k + 8*hh, col = lcol)
#pragma unroll
    for (int n = 0; n < N_; ++n) {
      const float* Xp =
          X + ((((size_t)n * CI_ + c) * H_) + h0 + 8 * hh) * H_ + w0 + lcol;
#pragma unroll
      for (int k = 0; k < 8; ++k)
        p[n][k] = fmaf(Xp[(size_t)k * H_], t[k], p[n][k]);
    }
  }

  // write P
#pragma unroll
  for (int n = 0; n < N_; ++n) {
    float* Pp =
        P + ((((size_t)n * O_ + o) * H_) + h0 + 8 * hh) * H_ + w0 + lcol;
#pragma unroll
    for (int k = 0; k < 8; ++k) Pp[(size_t)k * H_] = p[n][k];
  }
}

// ---------------------------------------------------------------------------
// Kernel 2: row sums rs[n,o,h] = sum_w P[n,o,h,w].  One wave32 per row,
// coalesced loads, butterfly shfl_xor reduction.
// ---------------------------------------------------------------------------
__global__ __launch_bounds__(256) void sylo_rowsum(const float* __restrict__ P,
                                                   float* __restrict__ rs) {
  const int row  = (int)((blockIdx.x * 256u + threadIdx.x) >> 5);  // 0..131071
  const int lane = threadIdx.x & 31;
  if (row >= N_ * O_ * H_) return;
  const float* rp = P + (size_t)row * H_;
  float acc = 0.0f;
  for (int w = lane; w < H_; w += 32) acc += rp[w];
#pragma unroll
  for (int m = 16; m >= 1; m >>= 1) acc += __shfl_xor(acc, m, 32);
  if (lane == 0) rs[row] = acc;
}

// ---------------------------------------------------------------------------
// Kernel 3: col sums cs[n,o,w] = sum_h P[n,o,h,w].  One block per (n,o),
// threads stride columns (fully coalesced per h iteration).
// ---------------------------------------------------------------------------
__global__ __launch_bounds__(256) void sylo_colsum(const float* __restrict__ P,
                                                   float* __restrict__ cs) {
  const int no = blockIdx.x;  // 0..N*O-1
  const float* base = P + (size_t)no * H_ * H_;
  for (int col = threadIdx.x; col < H_; col += 256) {
    float acc = 0.0f;
    for (int h = 0; h < H_; ++h) acc += base[(size_t)h * H_ + col];
    cs[(size_t)no * H_ + col] = acc;
  }
}

// ---------------------------------------------------------------------------
// Kernel 4: out = rs[h] + cs[w] - P + bias[o], diagonal zeroed. In-place,
// float4 vectorized along w.
// ---------------------------------------------------------------------------
__global__ __launch_bounds__(256) void sylo_final(float* __restrict__ P,
                                                  const float* __restrict__ rs,
                                                  const float* __restrict__ cs,
                                                  const float* __restrict__ bias) {
  const size_t idx = (size_t)blockIdx.x * 256 + threadIdx.x;  // N*O*H*(H/4)
  const int W4 = H_ / 4;
  const size_t total = (size_t)N_ * O_ * H_ * W4;
  if (idx >= total) return;
  const int w4 = (int)(idx % W4);
  const size_t r = idx / W4;
  const int h  = (int)(r % H_);
  const int no = (int)(r / H_);
  const int o  = no % O_;

  float4* pp = (float4*)(P + ((size_t)no * H_ + h) * H_) + w4;
  float4 v = *pp;
  const float rsv = rs[(size_t)no * H_ + h] + bias[o];
  const float* csp = cs + (size_t)no * H_ + w4 * 4;
  const int wbase = w4 * 4;
  float out0 = (h == wbase + 0) ? 0.0f : rsv + csp[0] - v.x;
  float out1 = (h == wbase + 1) ? 0.0f : rsv + csp[1] - v.y;
  float out2 = (h == wbase + 2) ? 0.0f : rsv + csp[2] - v.z;
  float out3 = (h == wbase + 3) ? 0.0f : rsv + csp[3] - v.w;
  *pp = make_float4(out0, out1, out2, out3);
}

extern "C" void kernel_launch(void* const* d_in, const int* in_sizes, int n_in,
                              void* d_out, int out_size, void* d_ws, size_t ws_size,
                              hipStream_t stream) {
  (void)in_sizes; (void)n_in; (void)out_size; (void)ws_size;
  const float* X    = (const float*)d_in[0];  // (N,CI,H,H)
  const float* L    = (const float*)d_in[1];  // (O,CI,H,R)
  const float* bias = (const float*)d_in[2];  // (O,)
  const float* S    = (const float*)d_in[3];  // (O,CI,R,1) signs

  float* P  = (float*)d_out;                  // (N,O,H,H), built in place
  float* rs = (float*)d_ws;                   // N*O*H floats
  float* cs = rs + (size_t)N_ * O_ * H_;      // N*O*H floats (1 MB total)

  // 4096 blocks: (O/8) x (H/16) x (H/16); 8 waves/block share the X tile.
  sylo_main<<<dim3(4096), dim3(256), 0, stream>>>(X, L, S, P);
  // one wave per row: N*O*H rows / 8 waves per block
  sylo_rowsum<<<dim3((N_ * O_ * H_) / 8), dim3(256), 0, stream>>>(P, rs);
  sylo_colsum<<<dim3(N_ * O_), dim3(256), 0, stream>>>(P, cs);
  // N*O*H*(H/4) float4 elements
  sylo_final<<<dim3((N_ * O_ * H_ * (H_ / 4)) / 256), dim3(256), 0, stream>>>(
      P, rs, cs, bias);
}